// DecoderWithNMS_3487513444546
// MI455X (gfx1250) — compile-verified
//
#include <hip/hip_runtime.h>
#include <hip/hip_bf16.h>
#include <cstdint>

// ---------------- problem constants ----------------
#define BATCH   8
#define WD      704
#define HD      800
#define PLANE   (WD * HD)          // 563200
#define NCH     9
#define KTOP    512
#define CONF_T  0.5f
#define MINY    (-40.0f)
#define MINZ    (-3.0f)
#define ZSIZE   4.0f
#define MEAN_H  1.52f
#define MEAN_W  1.63f
#define MEAN_L  3.88f

// ---------------- tuning ----------------
#define NBINS        8192          // 13-bit key histogram
#define CAND_CAP     1024
#define THREADS      512
#define SLICE        10240         // floats per block (PLANE = 55 * SLICE)
#define SLICES_PER_B (PLANE / SLICE)   // 55
#define CHUNK        2560          // floats per TDM chunk (4 chunks/slice)
#define NCHUNK       (SLICE / CHUNK)   // 4

// ---------------- workspace layout (bytes) ----------------
#define OFF_HIST     0
#define SZ_HIST      (BATCH * NBINS * 4)        // 262144
#define OFF_CANDCNT  (OFF_HIST + SZ_HIST)       // 262144
#define OFF_THR      (OFF_CANDCNT + BATCH * 4)  // 262176
#define OFF_CAND     (OFF_THR + BATCH * 4)      // 262208 (8B aligned)
#define SZ_CAND      (BATCH * CAND_CAP * 8)     // 65536
#define OFF_BOXES    (OFF_CAND + SZ_CAND)       // 327744
#define SZ_BOXES     (BATCH * KTOP * 8 * 4)     // 131072

typedef unsigned int v4u __attribute__((ext_vector_type(4)));
typedef int          v4i __attribute__((ext_vector_type(4)));
typedef int          v8i __attribute__((ext_vector_type(8)));
typedef float        v2f __attribute__((ext_vector_type(2)));
typedef float        v8f __attribute__((ext_vector_type(8)));

// order-preserving float -> u32 key (ascending)
__device__ __forceinline__ unsigned fkey(float f) {
    unsigned u = __float_as_uint(f);
    return (u & 0x80000000u) ? ~u : (u | 0x80000000u);
}

__device__ __forceinline__ float sigmoidf_(float x) { return 1.0f / (1.0f + expf(-x)); }

// ------------------------------------------------------------------
// K0: zero histograms + candidate counters
// ------------------------------------------------------------------
__global__ void k_zero(unsigned* __restrict__ p, int n) {
    int i = blockIdx.x * blockDim.x + threadIdx.x;
    if (i < n) p[i] = 0u;
}

// ------------------------------------------------------------------
// K1: stream conf plane via TDM (tensor_load_to_lds) -> key histogram
// grid: BATCH * SLICES_PER_B blocks of 512
// ------------------------------------------------------------------
__global__ __launch_bounds__(THREADS) void k_hist(const float* __restrict__ in,
                                                  unsigned* __restrict__ ghist) {
    __shared__ float    tbuf[2][CHUNK];      // 2 * 10240 B
    __shared__ unsigned hist[NBINS];         // 32768 B

    const int tid = threadIdx.x;
    const int b   = blockIdx.x / SLICES_PER_B;
    const int sl  = blockIdx.x % SLICES_PER_B;
    const long long base = (long long)b * NCH * PLANE + (long long)sl * SLICE;

    for (int i = tid; i < NBINS; i += THREADS) hist[i] = 0u;

    // issue a TDM load of chunk c into buffer buf (wave-uniform, wave 0 only)
    auto issue = [&](int c, int buf) {
        unsigned long long ga = (unsigned long long)(uintptr_t)(in + base + (long long)c * CHUNK);
        unsigned lds = (unsigned)(uintptr_t)(&tbuf[buf][0]);   // low 32 bits = LDS byte offset
        v4u g0;
        g0.x = 1u;                                        // count = 1 valid descriptor
        g0.y = lds;                                       // lds_addr (bytes)
        g0.z = (unsigned)(ga & 0xFFFFFFFFull);            // global_addr[31:0]
        g0.w = (unsigned)((ga >> 32) & 0x1FFFFFFull)      // global_addr[56:32]
             | (2u << 30);                                // type = 2 ("image")
        v8i g1;
        g1[0] = (int)(2u << 16);                          // data_size = 4B, mask/flags = 0
        g1[1] = (int)(((unsigned)CHUNK & 0xFFFFu) << 16); // tensor_dim0[15:0]
        g1[2] = (int)(((unsigned)CHUNK >> 16) | (1u << 16)); // dim0[31:16] | tensor_dim1 = 1
        g1[3] = (int)((unsigned)CHUNK << 16);             // tensor_dim1 hi | tile_dim0 = CHUNK
        g1[4] = 0;                                        // tile_dim1 = 0, tile_dim2 = 0
        g1[5] = (int)CHUNK;                               // tensor_dim0_stride lo
        g1[6] = 0;
        g1[7] = 0;
        v4i gz4 = {0, 0, 0, 0};
        v8i gz8 = {0, 0, 0, 0, 0, 0, 0, 0};
        __builtin_amdgcn_tensor_load_to_lds(g0, g1, gz4, gz4, gz8, 0);
    };

    if (tid < 32) issue(0, 0);

    for (int c = 0; c < NCHUNK; ++c) {
        if (tid < 32) {
            if (c + 1 < NCHUNK) {
                issue(c + 1, (c + 1) & 1);
                __builtin_amdgcn_s_wait_tensorcnt(1);   // chunk c landed
            } else {
                __builtin_amdgcn_s_wait_tensorcnt(0);
            }
        }
        __syncthreads();                                 // data visible to all waves
        const float* buf = &tbuf[c & 1][0];
        #pragma unroll
        for (int k = 0; k < CHUNK / THREADS; ++k) {
            float v = buf[tid + THREADS * k];
            atomicAdd(&hist[fkey(v) >> 19], 1u);         // top 13 bits
        }
        __syncthreads();                                 // done reading before buffer reuse
    }

    for (int i = tid; i < NBINS; i += THREADS) {
        unsigned h = hist[i];
        if (h) atomicAdd(&ghist[b * NBINS + i], h);
    }
}

// ------------------------------------------------------------------
// K2: per batch, find threshold bin T: largest bin with
//     sum_{bin >= T} count >= KTOP
// ------------------------------------------------------------------
__global__ __launch_bounds__(THREADS) void k_thresh(const unsigned* __restrict__ ghist,
                                                    int* __restrict__ thr) {
    __shared__ unsigned gsum[THREADS];
    const int b = blockIdx.x, tid = threadIdx.x;
    const unsigned* h = ghist + b * NBINS;
    unsigned s = 0;
    #pragma unroll
    for (int i = 0; i < NBINS / THREADS; ++i) s += h[tid * (NBINS / THREADS) + i];
    gsum[tid] = s;
    __syncthreads();
    if (tid == 0) {
        unsigned acc = 0;
        int T = 0;
        bool found = false;
        for (int g = THREADS - 1; g >= 0 && !found; --g) {
            unsigned gs = gsum[g];
            if (acc + gs >= (unsigned)KTOP) {
                unsigned a2 = acc;
                for (int bin = g * 16 + 15; bin >= g * 16; --bin) {
                    a2 += h[bin];
                    if (a2 >= (unsigned)KTOP) { T = bin; found = true; break; }
                }
            } else {
                acc += gs;
            }
        }
        thr[b] = T;
    }
}

// ------------------------------------------------------------------
// K3: second pass over conf plane, collect candidates with bin >= T
// packed as (key << 32) | ~idx  (u64 descending sort == exact top_k order)
// ------------------------------------------------------------------
__global__ __launch_bounds__(THREADS) void k_collect(const float* __restrict__ in,
                                                     const int* __restrict__ thr,
                                                     unsigned* __restrict__ candCnt,
                                                     unsigned long long* __restrict__ cand) {
    const int tid = threadIdx.x;
    const int b   = blockIdx.x / SLICES_PER_B;
    const int sl  = blockIdx.x % SLICES_PER_B;
    const long long s0 = (long long)sl * SLICE;
    const float4* pv = (const float4*)(in + (long long)b * NCH * PLANE + s0);
    const unsigned T = (unsigned)thr[b];

    #pragma unroll
    for (int k = 0; k < SLICE / 4 / THREADS; ++k) {       // 5 float4 per thread
        if (k + 1 < SLICE / 4 / THREADS)
            __builtin_prefetch(&pv[tid + THREADS * (k + 1)], 0, 0);
        float4 v = pv[tid + THREADS * k];
        unsigned sbase = (unsigned)s0 + 4u * (unsigned)(tid + THREADS * k);
        float vv[4] = {v.x, v.y, v.z, v.w};
        #pragma unroll
        for (int c = 0; c < 4; ++c) {
            unsigned key = fkey(vv[c]);
            if ((key >> 19) >= T) {
                unsigned pos = atomicAdd(&candCnt[b], 1u);
                if (pos < CAND_CAP)
                    cand[(long long)b * CAND_CAP + pos] =
                        ((unsigned long long)key << 32) | (unsigned)(~(sbase + c));
            }
        }
    }
}

// ------------------------------------------------------------------
// K4: per batch — bitonic sort candidates (descending), decode top 512
// ------------------------------------------------------------------
__global__ __launch_bounds__(THREADS) void k_sortdecode(const float* __restrict__ in,
                                                        const unsigned* __restrict__ candCnt,
                                                        const unsigned long long* __restrict__ cand,
                                                        float* __restrict__ boxes) {
    __shared__ unsigned long long keys[CAND_CAP];
    const int b = blockIdx.x, tid = threadIdx.x;
    unsigned cnt = candCnt[b];
    if (cnt > CAND_CAP) cnt = CAND_CAP;
    for (int t = tid; t < CAND_CAP; t += THREADS)
        keys[t] = (t < (int)cnt) ? cand[(long long)b * CAND_CAP + t] : 0ull;
    __syncthreads();

    // bitonic sort, descending
    for (int ksz = 2; ksz <= CAND_CAP; ksz <<= 1) {
        for (int j = ksz >> 1; j > 0; j >>= 1) {
            for (int t = tid; t < CAND_CAP; t += THREADS) {
                int p = t ^ j;
                if (p > t) {
                    bool up = ((t & ksz) == 0);
                    unsigned long long a = keys[t], c = keys[p];
                    bool sw = up ? (a < c) : (a > c);
                    if (sw) { keys[t] = c; keys[p] = a; }
                }
            }
            __syncthreads();
        }
    }

    if (tid < KTOP) {
        unsigned long long sk = keys[tid];
        unsigned idx = ~(unsigned)(sk & 0xFFFFFFFFull);
        if (idx >= (unsigned)PLANE) idx = 0;              // defensive (shouldn't occur)
        const float* p0 = in + (long long)b * NCH * PLANE + idx;
        float o0 = p0[0 * PLANE], o1 = p0[1 * PLANE], o2 = p0[2 * PLANE];
        float o3 = p0[3 * PLANE], o4 = p0[4 * PLANE], o5 = p0[5 * PLANE];
        float o6 = p0[6 * PLANE], o7 = p0[7 * PLANE], o8 = p0[8 * PLANE];
        float gx = (float)(idx / HD);
        float gy = (float)(idx % HD);
        float* o = boxes + ((long long)b * KTOP + tid) * 8;
        o[0] = sigmoidf_(o0);
        o[1] = sigmoidf_(o1) + gx;
        o[2] = sigmoidf_(o2) + gy + MINY;
        o[3] = sigmoidf_(o3) * ZSIZE + MINZ;
        o[4] = expf(o4) * MEAN_H;
        o[5] = expf(o5) * MEAN_W;
        o[6] = expf(o6) * MEAN_L;
        o[7] = atan2f(tanhf(o7), tanhf(o8));
    }
}

// ------------------------------------------------------------------
// K5: per batch — suppression bit-matrix (WMMA outer-sum for vol_i+vol_j),
//     division-free IoU test, barrier-free single-wave greedy pass
// ------------------------------------------------------------------
__global__ __launch_bounds__(THREADS) void k_nms(const float* __restrict__ boxes,
                                                 float* __restrict__ outp) {
    __shared__ float X1[KTOP], X2[KTOP], Y1[KTOP], Y2[KTOP];
    __shared__ float Z1[KTOP], Z2[KTOP], VOL[KTOP], CF[KTOP];
    __shared__ unsigned sup[KTOP * 16];    // 32 KB: bit j of row i
    __shared__ unsigned keepW[16];

    const int b = blockIdx.x, tid = threadIdx.x;
    const int wv = tid >> 5, lane = tid & 31;

    const float* bx = boxes + ((long long)b * KTOP + tid) * 8;
    float cf = bx[0], x = bx[1], y = bx[2], z = bx[3];
    float h = bx[4], w = bx[5], l = bx[6];
    X1[tid] = x - l * 0.5f;  X2[tid] = x + l * 0.5f;
    Y1[tid] = y - w * 0.5f;  Y2[tid] = y + w * 0.5f;
    Z1[tid] = z - h * 0.5f;  Z2[tid] = z + h * 0.5f;
    VOL[tid] = l * w * h;    CF[tid] = cf;
    #pragma unroll
    for (int k = 0; k < 16; ++k) sup[tid * 16 + k] = 0u;
    __syncthreads();

    // 528 upper-triangular 16x16 tiles, one wave per tile
    for (int t = wv; t < 528; t += 16) {
        int ti = 0, rem = t;
        while (rem >= (32 - ti)) { rem -= (32 - ti); ++ti; }
        int tj = ti + rem;
        int i0 = ti << 4, j0 = tj << 4;

        // D[i][j] = vol_i * 1 + 1 * vol_j  (exact fp32 outer sum via WMMA)
        v2f a, bb;
        if (lane < 16) { a.x = VOL[i0 + lane]; a.y = 1.0f; bb.x = 1.0f; bb.y = VOL[j0 + lane]; }
        else           { a.x = 0.0f;           a.y = 0.0f; bb.x = 0.0f; bb.y = 0.0f; }
        v8f c = {};
        c = __builtin_amdgcn_wmma_f32_16x16x4_f32(false, a, false, bb,
                                                  (short)0, c, false, false);
        int i_base = i0 + ((lane >> 4) << 3);
        int j = j0 + (lane & 15);
        #pragma unroll
        for (int r = 0; r < 8; ++r) {
            int i = i_base + r;
            if (j > i) {
                float ox = fmaxf(fminf(X2[i], X2[j]) - fmaxf(X1[i], X1[j]), 0.0f);
                float oy = fmaxf(fminf(Y2[i], Y2[j]) - fmaxf(Y1[i], Y1[j]), 0.0f);
                float oz = fmaxf(fminf(Z2[i], Z2[j]) - fmaxf(Z1[i], Z1[j]), 0.0f);
                float ov = ox * oy * oz;
                // iou > CONF  <=>  ov > CONF * (vol_i + vol_j - ov + eps)   (denom > 0)
                if (ov > CONF_T * (c[r] - ov + 1e-6f))
                    atomicOr(&sup[i * 16 + (j >> 5)], 1u << (j & 31));
            }
        }
    }

    // initial keep = conf > CONF  (wave32 ballot)
    unsigned long long bal = __ballot(CF[tid] > CONF_T);
    if (lane == 0) keepW[wv] = (unsigned)bal;
    __syncthreads();

    // greedy pass: wave 0 only, keep bits register-resident (lane k owns word k),
    // alive-bit broadcast via shuffle -> zero block barriers in the serial chain
    if (wv == 0) {
        unsigned keepReg = (lane < 16) ? keepW[lane] : 0u;
        for (int i = 0; i < KTOP; ++i) {
            unsigned kw = __shfl(keepReg, i >> 5);
            if ((kw >> (i & 31)) & 1u)                    // wave-uniform branch
                keepReg &= ~sup[i * 16 + (lane & 15)];
        }
        if (lane < 16) keepW[lane] = keepReg;
    }
    __syncthreads();

    float kf = ((keepW[tid >> 5] >> (tid & 31)) & 1u) ? 1.0f : 0.0f;
    float* o = outp + ((long long)b * KTOP + tid) * 8;
    #pragma unroll
    for (int c2 = 0; c2 < 8; ++c2) o[c2] = bx[c2] * kf;
}

// ------------------------------------------------------------------
extern "C" void kernel_launch(void* const* d_in, const int* in_sizes, int n_in,
                              void* d_out, int out_size, void* d_ws, size_t ws_size,
                              hipStream_t stream) {
    (void)in_sizes; (void)n_in; (void)out_size; (void)ws_size;
    const float* in  = (const float*)d_in[0];
    float*       out = (float*)d_out;
    char*        ws  = (char*)d_ws;

    unsigned*            hist    = (unsigned*)(ws + OFF_HIST);
    unsigned*            candCnt = (unsigned*)(ws + OFF_CANDCNT);
    int*                 thr     = (int*)(ws + OFF_THR);
    unsigned long long*  cand    = (unsigned long long*)(ws + OFF_CAND);
    float*               boxes   = (float*)(ws + OFF_BOXES);

    const int nz = (SZ_HIST + BATCH * 4) / 4;            // hist + candCnt words
    k_zero<<<(nz + THREADS - 1) / THREADS, THREADS, 0, stream>>>((unsigned*)(ws + OFF_HIST), nz);

    k_hist<<<BATCH * SLICES_PER_B, THREADS, 0, stream>>>(in, hist);
    k_thresh<<<BATCH, THREADS, 0, stream>>>(hist, thr);
    k_collect<<<BATCH * SLICES_PER_B, THREADS, 0, stream>>>(in, thr, candCnt, cand);
    k_sortdecode<<<BATCH, THREADS, 0, stream>>>(in, candCnt, cand, boxes);
    k_nms<<<BATCH, THREADS, 0, stream>>>(boxes, out);
}